// StochasticSubSampler_45131516346504
// MI455X (gfx1250) — compile-verified
//
#include <hip/hip_runtime.h>
#include <hip/hip_bf16.h>

// ---------------------------------------------------------------------------
// StochasticSubSampler backward-hook update for MI455X (gfx1250, wave32).
// Dense 64MB f32 accumulator (L2-resident on 192MB L2) + atomic scatter,
// 2-level 12+12-bit radix-select with WMMA-assisted histogram reduction,
// ballot-aggregated compaction, LDS bitonic tie-break, and a launch-fused
// bitonic sort (LDS presort + LDS-fused merge stages: 55 launches vs 210).
// ---------------------------------------------------------------------------

#define CDIM   4096
#define NUMEL  (CDIM * CDIM)          // 16,777,216
#define S_N    3355443
#define R_N    1048576
#define TIE_CAP 32768
#define TIE_LDS 4096
#define SEG    2048                    // LDS bitonic segment (16KB of u64)
// adj_decay = 1 - (S/NUMEL) * (1 - 0.95)
#define ADJ_DECAY (1.0f - ((float)S_N / (float)NUMEL) * (1.0f - 0.95f))

typedef __attribute__((ext_vector_type(2))) float v2f;
typedef __attribute__((ext_vector_type(8))) float v8f;
typedef unsigned long long u64;

// --- WMMA 64-element sum: D = ones(16x4) x B(4x16); column sums of B land in
// D[.][j]; lanes 0..15 hold N=0..15, so a 16-lane xor-reduce of d[0] yields
// the exact total of all 64 f32 values (exact for integer counts <= 2^24).
// Layout-robust: the total is invariant to the exact K/N striping of A and B.
__device__ __forceinline__ float wave_sum64_wmma(const float* v) {
  const int lane = (int)(threadIdx.x & 31u);
  v2f a; a.x = 1.0f; a.y = 1.0f;                   // ones A (16x4)
  v2f b; b.x = v[2 * lane]; b.y = v[2 * lane + 1]; // 64 data values per wave
  v8f c = {};
  v8f d = __builtin_amdgcn_wmma_f32_16x16x4_f32(
      /*neg_a=*/false, a, /*neg_b=*/false, b,
      /*c_mod=*/(short)0, c, /*reuse_a=*/false, /*reuse_b=*/false);
  float s = d[0];
  s += __shfl_xor(s, 1);
  s += __shfl_xor(s, 2);
  s += __shfl_xor(s, 4);
  s += __shfl_xor(s, 8);
  return s;   // full 64-element sum in every lane
}

// ---------------------------------------------------------------------------
__global__ void __launch_bounds__(256) k_init(float4* __restrict__ acc4,
                                              unsigned* __restrict__ h1,
                                              unsigned* __restrict__ h2,
                                              int* __restrict__ prm) {
  const int n4 = NUMEL / 4;
  const float4 z = make_float4(0.f, 0.f, 0.f, 0.f);
  for (int i = blockIdx.x * blockDim.x + threadIdx.x; i < n4;
       i += gridDim.x * blockDim.x)
    acc4[i] = z;
  int t = blockIdx.x * blockDim.x + threadIdx.x;
  if (t < 4096) { h1[t] = 0u; h2[t] = 0u; }
  if (t < 16) prm[t] = 0;
}

__global__ void __launch_bounds__(256) k_scatter_grad(
    const int* __restrict__ sidx, const float* __restrict__ grad,
    float* __restrict__ acc) {
  int i = blockIdx.x * blockDim.x + threadIdx.x;
  if (i < S_N) atomicAdd(&acc[(unsigned)sidx[i]], fabsf(grad[i]));
}

__global__ void __launch_bounds__(256) k_scatter_res(
    const int* __restrict__ ridx, const float* __restrict__ rval,
    float* __restrict__ acc) {
  int i = blockIdx.x * blockDim.x + threadIdx.x;
  if (i < R_N) {
    int r = ridx[2 * i], c = ridx[2 * i + 1];
    atomicAdd(&acc[r * CDIM + c], ADJ_DECAY * rval[i]);
  }
}

// Pass-1 histogram: 12 MSBs of the f32 key (values >= 0 => bits monotonic).
__global__ void __launch_bounds__(256) k_hist1(const float4* __restrict__ acc4,
                                               unsigned* __restrict__ hist) {
  __shared__ unsigned sh[4096];
  for (int t = threadIdx.x; t < 4096; t += 256) sh[t] = 0u;
  __syncthreads();
  const int n4 = NUMEL / 4;
  for (int i = blockIdx.x * blockDim.x + threadIdx.x; i < n4;
       i += gridDim.x * blockDim.x) {
    __builtin_prefetch(acc4 + i + 2048, 0, 0);   // global_prefetch_b8
    float4 v = acc4[i];
    atomicAdd(&sh[__float_as_uint(v.x) >> 20], 1u);
    atomicAdd(&sh[__float_as_uint(v.y) >> 20], 1u);
    atomicAdd(&sh[__float_as_uint(v.z) >> 20], 1u);
    atomicAdd(&sh[__float_as_uint(v.w) >> 20], 1u);
  }
  __syncthreads();
  for (int t = threadIdx.x; t < 4096; t += 256) {
    unsigned cnt = sh[t];
    if (cnt) atomicAdd(&hist[t], cnt);
  }
}

// Pass-2 histogram: next 12 bits, restricted to the selected top bin b1.
__global__ void __launch_bounds__(256) k_hist2(const float4* __restrict__ acc4,
                                               unsigned* __restrict__ hist,
                                               const int* prm) {
  __shared__ unsigned sh[4096];
  for (int t = threadIdx.x; t < 4096; t += 256) sh[t] = 0u;
  __syncthreads();
  const unsigned b1 = (unsigned)prm[0];
  const int n4 = NUMEL / 4;
  for (int i = blockIdx.x * blockDim.x + threadIdx.x; i < n4;
       i += gridDim.x * blockDim.x) {
    __builtin_prefetch(acc4 + i + 2048, 0, 0);
    float4 v = acc4[i];
    unsigned k;
    k = __float_as_uint(v.x); if ((k >> 20) == b1) atomicAdd(&sh[(k >> 8) & 0xFFFu], 1u);
    k = __float_as_uint(v.y); if ((k >> 20) == b1) atomicAdd(&sh[(k >> 8) & 0xFFFu], 1u);
    k = __float_as_uint(v.z); if ((k >> 20) == b1) atomicAdd(&sh[(k >> 8) & 0xFFFu], 1u);
    k = __float_as_uint(v.w); if ((k >> 20) == b1) atomicAdd(&sh[(k >> 8) & 0xFFFu], 1u);
  }
  __syncthreads();
  for (int t = threadIdx.x; t < 4096; t += 256) {
    unsigned cnt = sh[t];
    if (cnt) atomicAdd(&hist[t], cnt);
  }
}

// Threshold-bin select over a 4096-bin histogram. Group (64-bin) partial sums
// are computed on the matrix unit (exact: counts <= 2^24 in f32), then one
// lane walks 64 group sums + 64 bins from the top to find the crossing bin.
__global__ void __launch_bounds__(256) k_select(const unsigned* __restrict__ hist,
                                                int* prm, int pass) {
  __shared__ float shf[4096];
  __shared__ float gsum[64];
  for (int t = threadIdx.x; t < 4096; t += 256) shf[t] = (float)hist[t];
  __syncthreads();
  const int wave = threadIdx.x >> 5;
  for (int g8 = 0; g8 < 8; ++g8) {          // uniform: EXEC all-1s at WMMA
    int g = wave * 8 + g8;
    float s = wave_sum64_wmma(&shf[g * 64]);
    if ((threadIdx.x & 31u) == 0u) gsum[g] = s;
  }
  __syncthreads();
  if (threadIdx.x == 0) {
    unsigned target = (pass == 0) ? (unsigned)R_N : (unsigned)(R_N - prm[1]);
    unsigned cum = 0;
    int bsel = 0;
    for (int g = 63; g >= 0; --g) {
      unsigned gs = (unsigned)(gsum[g] + 0.5f);
      if (cum + gs >= target) {
        for (int b = g * 64 + 63;; --b) {
          unsigned hb = hist[b];
          if (cum + hb >= target || b == g * 64) { bsel = b; break; }
          cum += hb;
        }
        break;
      }
      cum += gs;
    }
    if (pass == 0) { prm[0] = bsel; prm[1] = (int)cum; }       // b1, count>bin
    else { prm[2] = (prm[0] << 12) | bsel; prm[3] = prm[1] + (int)cum; }
  }
}

// Compact entries above the 24-bit threshold prefix. Wave32 ballot-aggregated
// global atomics (one atomic per wave instead of per lane). Ties (== prefix)
// go to a side buffer for exact tie-breaking.
__global__ void __launch_bounds__(256) k_compact(const float* __restrict__ acc,
                                                 const int* prm,
                                                 u64* __restrict__ mainv,
                                                 u64* __restrict__ tiev,
                                                 int* cnts) {
  const unsigned P = (unsigned)prm[2];
  const int lane = (int)(threadIdx.x & 31u);
  for (int i = blockIdx.x * blockDim.x + threadIdx.x; i < NUMEL;
       i += gridDim.x * blockDim.x) {
    unsigned key = __float_as_uint(acc[i]);
    unsigned p = key >> 8;
    bool isMain = p > P;
    u64 m = __ballot(isMain);                 // wave32: bits [31:0]
    if (m) {
      int leader = __ffsll(m) - 1;
      int base = 0;
      if (lane == leader) base = atomicAdd(&cnts[0], (int)__popcll(m));
      base = __shfl(base, leader);
      if (isMain) {
        int pos = base + (int)__popcll(m & ((1ull << lane) - 1ull));
        if (pos < R_N) mainv[pos] = ((u64)(~key) << 32) | (unsigned)i;
      }
    }
    if (p == P) {
      int pos = atomicAdd(&cnts[1], 1);
      if (pos < TIE_CAP) tiev[pos] = ((u64)(~key) << 32) | (unsigned)i;
    }
  }
}

// Sort ties (shared 24-bit prefix) ascending on (~key, idx) in LDS and append
// the needed R - cnt_main entries after the strict-greater block.
__global__ void __launch_bounds__(1024) k_tiesort(u64* __restrict__ mainv,
                                                  const u64* __restrict__ tiev,
                                                  const int* prm) {
  __shared__ u64 s[TIE_LDS];
  int nt = prm[5];
  if (nt > TIE_CAP) nt = TIE_CAP;
  if (nt > TIE_LDS) nt = TIE_LDS;
  for (int i = threadIdx.x; i < TIE_LDS; i += 1024)
    s[i] = (i < nt) ? tiev[i] : ~0ull;
  __syncthreads();
  for (int k = 2; k <= TIE_LDS; k <<= 1) {
    for (int j = k >> 1; j > 0; j >>= 1) {
      for (int i = threadIdx.x; i < TIE_LDS; i += 1024) {
        int ixj = i ^ j;
        if (ixj > i) {
          u64 x = s[i], y = s[ixj];
          bool up = ((i & k) == 0);
          if ((x > y) == up) { s[i] = y; s[ixj] = x; }
        }
      }
      __syncthreads();
    }
  }
  int g = prm[4];
  if (g > R_N) g = R_N;
  int need = R_N - g;
  for (int i = threadIdx.x; i < need; i += 1024)
    mainv[g + i] = (i < TIE_LDS) ? s[i] : ~0ull;
}

// --- Bitonic sort over R_N packed (~key, idx) u64s, ascending == value
// descending with ascending index on exact value ties. Stages fused into LDS
// wherever the compare distance fits a 2048-element segment.

// Full local network k=2..SEG. Direction bits come from the GLOBAL index, so
// this is bit-identical to the flat network (segment base is a multiple of
// SEG, so all direction bits for k<=SEG are locally determined).
__global__ void __launch_bounds__(1024) k_bitonic_presort(u64* __restrict__ a) {
  __shared__ u64 s[SEG];
  const unsigned base = blockIdx.x * (unsigned)SEG;
  for (int i = threadIdx.x; i < SEG; i += 1024) s[i] = a[base + i];
  __syncthreads();
  for (unsigned k = 2; k <= (unsigned)SEG; k <<= 1) {
    for (unsigned j = k >> 1; j > 0; j >>= 1) {
      for (int i = threadIdx.x; i < SEG; i += 1024) {
        unsigned ixj = (unsigned)i ^ j;
        if (ixj > (unsigned)i) {
          bool up = (((base + (unsigned)i) & k) == 0u);
          u64 x = s[i], y = s[ixj];
          if ((x > y) == up) { s[i] = y; s[ixj] = x; }
        }
      }
      __syncthreads();
    }
  }
  for (int i = threadIdx.x; i < SEG; i += 1024) a[base + i] = s[i];
}

// One global stage (compare distance j >= SEG).
__global__ void __launch_bounds__(256) k_bitonic(u64* __restrict__ a,
                                                 unsigned j, unsigned k) {
  unsigned i = blockIdx.x * blockDim.x + threadIdx.x;
  unsigned ixj = i ^ j;
  if (ixj > i && ixj < (unsigned)R_N) {
    u64 x = a[i], y = a[ixj];
    bool up = ((i & k) == 0u);
    if ((x > y) == up) { a[i] = y; a[ixj] = x; }
  }
}

// Fused tail stages j = SEG/2 .. 1 for a given k >= 2*SEG. Direction is
// uniform per segment: (base & k) since k exceeds the segment span.
__global__ void __launch_bounds__(1024) k_bitonic_merge_local(
    u64* __restrict__ a, unsigned k) {
  __shared__ u64 s[SEG];
  const unsigned base = blockIdx.x * (unsigned)SEG;
  const bool up = ((base & k) == 0u);
  for (int i = threadIdx.x; i < SEG; i += 1024) s[i] = a[base + i];
  __syncthreads();
  for (unsigned j = (unsigned)SEG >> 1; j > 0; j >>= 1) {
    for (int i = threadIdx.x; i < SEG; i += 1024) {
      unsigned ixj = (unsigned)i ^ j;
      if (ixj > (unsigned)i) {
        u64 x = s[i], y = s[ixj];
        if ((x > y) == up) { s[i] = y; s[ixj] = x; }
      }
    }
    __syncthreads();
  }
  for (int i = threadIdx.x; i < SEG; i += 1024) a[base + i] = s[i];
}

__global__ void __launch_bounds__(256) k_output(const u64* __restrict__ mainv,
                                                int* __restrict__ outIdx,
                                                float* __restrict__ outVal) {
  int i = blockIdx.x * blockDim.x + threadIdx.x;
  if (i < R_N) {
    u64 u = mainv[i];
    unsigned key = ~(unsigned)(u >> 32);
    unsigned idx = (unsigned)u;
    outIdx[2 * i]     = (int)(idx >> 12);         // row = idx / 4096
    outIdx[2 * i + 1] = (int)(idx & (CDIM - 1));  // col = idx % 4096
    outVal[i] = __uint_as_float(key);
  }
}

// ---------------------------------------------------------------------------
extern "C" void kernel_launch(void* const* d_in, const int* in_sizes, int n_in,
                              void* d_out, int out_size, void* d_ws,
                              size_t ws_size, hipStream_t stream) {
  (void)in_sizes; (void)n_in; (void)out_size; (void)ws_size;
  const int*   sidx = (const int*)d_in[0];    // sample_index [S]
  const int*   ridx = (const int*)d_in[1];    // result_index [R,2]
  const float* rval = (const float*)d_in[2];  // result_value [R]
  const float* grad = (const float*)d_in[3];  // gradient     [S]

  // Workspace layout (~76 MB): acc 64MB | hist1 16KB | hist2 16KB | params |
  // main 8MB | tie 256KB
  char* w = (char*)d_ws;
  float*    acc = (float*)w;
  unsigned* h1  = (unsigned*)(w + (size_t)NUMEL * 4);
  unsigned* h2  = h1 + 4096;
  int*      prm = (int*)(h2 + 4096);
  u64* mainv = (u64*)(w + (size_t)NUMEL * 4 + 65536);
  u64* tiev  = mainv + R_N;

  k_init<<<8192, 256, 0, stream>>>((float4*)acc, h1, h2, prm);
  k_scatter_grad<<<(S_N + 255) / 256, 256, 0, stream>>>(sidx, grad, acc);
  k_scatter_res<<<(R_N + 255) / 256, 256, 0, stream>>>(ridx, rval, acc);
  k_hist1<<<1024, 256, 0, stream>>>((const float4*)acc, h1);
  k_select<<<1, 256, 0, stream>>>(h1, prm, 0);
  k_hist2<<<1024, 256, 0, stream>>>((const float4*)acc, h2, prm);
  k_select<<<1, 256, 0, stream>>>(h2, prm, 1);
  k_compact<<<2048, 256, 0, stream>>>(acc, prm, mainv, tiev, prm + 4);
  k_tiesort<<<1, 1024, 0, stream>>>(mainv, tiev, prm);

  // Launch-fused bitonic sort: 1 presort + per-k (global j>=SEG stages + one
  // LDS-fused tail) = 55 launches total (vs 210 flat).
  k_bitonic_presort<<<R_N / SEG, 1024, 0, stream>>>(mainv);
  for (unsigned k = 2 * SEG; k <= (unsigned)R_N; k <<= 1) {
    for (unsigned j = k >> 1; j >= (unsigned)SEG; j >>= 1)
      k_bitonic<<<R_N / 256, 256, 0, stream>>>(mainv, j, k);
    k_bitonic_merge_local<<<R_N / SEG, 1024, 0, stream>>>(mainv, k);
  }

  k_output<<<R_N / 256, 256, 0, stream>>>(mainv, (int*)d_out,
                                          ((float*)d_out) + 2 * (size_t)R_N);
}